// BERT_CRF_73847667688014
// MI455X (gfx1250) — compile-verified
//
#include <hip/hip_runtime.h>
#include <math.h>

// CRF forward (log partition): fv'[n] = emit[n] + logsumexp_p(fv[p] + trans[n,p]).
// exp(fv+trans) = exp(trans)*exp(fv): per-step 64x64x16 f32 GEMM S = E x P using
// V_WMMA_F32_16X16X4_F32, with E = exp(trans) precomputed once. Forward vars live
// in registers (each lane owns 8 (next,batch) entries); LDS only carries P and a
// tiny 4x16 partial-max array per step. All B-fragments are preloaded so the 16
// WMMAs issue back-to-back; fv updates are branchless (v_cndmask).
#define TAGSET 50
#define TT (TAGSET + 2)        // 52 tags incl START/STOP
#define START_TAG TAGSET       // 50
#define STOP_TAG (TAGSET + 1)  // 51
#define TP 64                  // padded tag dim
#define BG 16                  // batch group = WMMA N dim
#define NEG_INF_VAL (-10000.0f)

typedef __attribute__((ext_vector_type(2))) float v2f;
typedef __attribute__((ext_vector_type(4))) float v4f;
typedef __attribute__((ext_vector_type(8))) float v8f;

#if defined(__has_builtin) && __has_builtin(__builtin_amdgcn_logf)
// raw v_log_f32 (log2), no denormal fixup; denormal sums -> -inf (harmless here)
__device__ __forceinline__ float fast_log(float x) {
    return __builtin_amdgcn_logf(x) * 0.6931471805599453f;
}
#else
__device__ __forceinline__ float fast_log(float x) { return __logf(x); }
#endif

// ---- Kernel 1: Epad[n][p] = exp(trans[n][p]) on 64x64, zero outside TxT ----
__global__ void crf_build_exp_trans(const float* __restrict__ trans,
                                    float* __restrict__ Epad) {
    int idx = blockIdx.x * blockDim.x + threadIdx.x;
    if (idx >= TP * TP) return;
    int n = idx >> 6, p = idx & 63;
    float v = 0.0f;
    if (n < TT && p < TT) {
        v = expf(trans[n * TT + p]);   // exp(-10000) underflows to exact 0
    }
    Epad[idx] = v;
}

// ---- Kernel 2: forward recurrence. 1 block = 16 sequences, 4 waves. ----
__global__ __launch_bounds__(128)
void crf_forward(const float* __restrict__ scores,   // (B, L, TT)
                 const float* __restrict__ trans,    // (TT, TT)
                 const int*   __restrict__ lengths,  // (B,)
                 const float* __restrict__ Epad,     // (64, 64)
                 float* __restrict__ out,            // (B,)
                 int Lt)
{
    __shared__ float pLDS [TP * BG];   // exp(fv - m), [prev][batch]
    __shared__ float mPart[4 * BG];    // per-wave partial max, [wave][batch]

    const int tid  = threadIdx.x;
    const int wave = tid >> 5;         // owns next rows [16w, 16w+16)
    const int lane = tid & 31;
    const int half = lane >> 4;        // 0/1
    const int l16  = lane & 15;
    const int b    = l16;              // batch within group
    const int bG0  = blockIdx.x * BG;
    const int nxBase = 16 * wave + 8 * half;   // lane's 8 next rows: nxBase..nxBase+7
    const int koff = half * 2;

    // ---- Preload A fragments (rows of E for this wave's M tile), loop-invariant.
    // A 16x4 f32 layout: lanes 0-15 M=lane {K=0,K=1}; lanes 16-31 M=lane-16 {K=2,K=3}.
    const int mrow = 16 * wave + l16;
    v2f aF[16];
    #pragma unroll
    for (int k = 0; k < 16; ++k) {
        aF[k].x = Epad[mrow * TP + 4 * k + koff + 0];
        aF[k].y = Epad[mrow * TP + 4 * k + koff + 1];
    }

    // ---- fv in registers: fv[j] is forward var for (next = nxBase+j, batch = b).
    float fv[8];
    #pragma unroll
    for (int j = 0; j < 8; ++j) {
        int nx = nxBase + j;
        fv[j] = (nx == START_TAG) ? 0.0f : ((nx < TT) ? NEG_INF_VAL : -1.0e30f);
    }

    // group's max length + this lane's length
    int maxLen = 1;
    #pragma unroll
    for (int j = 0; j < BG; ++j) maxLen = max(maxLen, lengths[bG0 + j]);
    const int myLen = lengths[bG0 + b];

    const float* emitBase = scores + (size_t)(bG0 + b) * (size_t)Lt * TT + nxBase;
    const bool e0ok = (nxBase + 3 < TT);   // lane-invariant across steps
    const bool e1ok = (nxBase + 7 < TT);

    for (int step = 0; step < maxLen; ++step) {
        // ---- per-wave partial max over this wave's 16 rows for batch b (registers only)
        float wmax = fv[0];
        #pragma unroll
        for (int j = 1; j < 8; ++j) wmax = fmaxf(wmax, fv[j]);
        wmax = fmaxf(wmax, __shfl_xor(wmax, 16, 32));
        if (half == 0) mPart[wave * BG + b] = wmax;

        __syncthreads();   // mPart ready; previous step's pLDS readers are done

        float mB = fmaxf(fmaxf(mPart[0 * BG + b], mPart[1 * BG + b]),
                         fmaxf(mPart[2 * BG + b], mPart[3 * BG + b]));

        // ---- P rows owned by this lane, straight from registers
        #pragma unroll
        for (int j = 0; j < 8; ++j)
            pLDS[(nxBase + j) * BG + b] = __expf(fv[j] - mB);

        __syncthreads();   // P ready

        // ---- emit loads issued before WMMA so global latency hides under math.
        // Lane's 8 next indices are consecutive & 16B-aligned (T = 52 ≡ 0 mod 4).
        float ee[8];
        #pragma unroll
        for (int j = 0; j < 8; ++j) ee[j] = 0.0f;
        const float* emitRow = emitBase + (size_t)step * TT;
        if (e0ok) { v4f e0 = *(const v4f*)(emitRow);
                    ee[0]=e0.x; ee[1]=e0.y; ee[2]=e0.z; ee[3]=e0.w; }
        if (e1ok) { v4f e1 = *(const v4f*)(emitRow + 4);
                    ee[4]=e1.x; ee[5]=e1.y; ee[6]=e1.z; ee[7]=e1.w; }

        // ---- Preload ALL 16 B fragments (32 VGPRs) so the ds_loads pipeline,
        // then issue the 16 WMMAs back-to-back on two independent chains.
        // B 4x16 f32 layout: lanes 0-15 {K=0,K=1}, lanes 16-31 {K=2,K=3}, N=lane%16.
        v2f bF[16];
        #pragma unroll
        for (int k = 0; k < 16; ++k) {
            bF[k].x = pLDS[(4 * k + koff + 0) * BG + b];
            bF[k].y = pLDS[(4 * k + koff + 1) * BG + b];
        }

        v8f c0 = {}, c1 = {};
        #pragma unroll
        for (int k = 0; k < 8; ++k) {
            c0 = __builtin_amdgcn_wmma_f32_16x16x4_f32(
                     false, aF[k],     false, bF[k],     (short)0, c0, false, false);
            c1 = __builtin_amdgcn_wmma_f32_16x16x4_f32(
                     false, aF[k + 8], false, bF[k + 8], (short)0, c1, false, false);
        }

        // ---- fv update, branchless: compute then select (v_cndmask).
        // c[j] = S[next = nxBase + j][batch = b]; log(0) -> -inf; garbage lanes masked.
        const bool act = (step < myLen);
        #pragma unroll
        for (int j = 0; j < 8; ++j) {
            float nv = ee[j] + mB + fast_log(c0[j] + c1[j]);
            bool upd = ((nxBase + j) < TT) && act;
            fv[j] = upd ? nv : fv[j];
        }
    }

    // ---- publish final fv to LDS (reuse pLDS), then terminal logsumexp.
    __syncthreads();   // last step's pLDS readers done
    #pragma unroll
    for (int j = 0; j < 8; ++j)
        pLDS[(nxBase + j) * BG + b] = fv[j];
    __syncthreads();

    if (wave == 0 && lane < BG) {
        float m2 = -3.0e38f;
        for (int n = 0; n < TT; ++n) {
            float v = pLDS[n * BG + lane] + trans[STOP_TAG * TT + n];
            m2 = fmaxf(m2, v);
        }
        float s2 = 0.0f;
        for (int n = 0; n < TT; ++n) {
            float v = pLDS[n * BG + lane] + trans[STOP_TAG * TT + n];
            s2 += expf(v - m2);
        }
        out[bG0 + lane] = m2 + logf(s2);
    }
}

extern "C" void kernel_launch(void* const* d_in, const int* in_sizes, int n_in,
                              void* d_out, int out_size, void* d_ws, size_t ws_size,
                              hipStream_t stream) {
    const float* scores  = (const float*)d_in[0];   // (B, L, T) f32
    const float* trans   = (const float*)d_in[1];   // (T, T)    f32
    const int*   lengths = (const int*)  d_in[2];   // (B,)      int
    float*       out     = (float*)d_out;           // (B,)      f32
    float*       Epad    = (float*)d_ws;            // 64*64*4 = 16 KB scratch

    const int B = in_sizes[2];
    const int L = in_sizes[0] / (B * TT);

    crf_build_exp_trans<<<(TP * TP + 255) / 256, 256, 0, stream>>>(trans, Epad);
    crf_forward<<<B / BG, 128, 0, stream>>>(scores, trans, lengths, Epad, out, L);
}